// NonLinearDiffusion_33217277067763
// MI455X (gfx1250) — compile-verified
//
#include <hip/hip_runtime.h>
#include <cstdint>

#define TILE_X 64
#define TILE_Y 32
#define IN_W      (TILE_X + 4)   // 68 valid columns
#define IN_H      (TILE_Y + 4)   // 36 rows
#define IN_STRIDE 72             // padded row stride (multiple of 4 floats -> 16B)
#define FL_W      (TILE_X + 2)   // 66
#define FL_H      (TILE_Y + 2)   // 34
#define NTHREADS  256

// input-local (row, lx) -> LDS index; lx in [0,67], layout shifted by +2 so the
// 64-float middle segment (lx=2..65) lands 16B-aligned and halos 8B-aligned.
#define SIN_IDX(row, lx) ((row) * IN_STRIDE + (lx) + 2)

// ---- CDNA5 async global->LDS copies (gfx1250), tracked by ASYNCcnt ----
__device__ __forceinline__ void async_ld_b128(unsigned lds_off, const float* gaddr) {
    asm volatile("global_load_async_to_lds_b128 %0, %1, off"
                 :: "v"(lds_off), "v"(gaddr) : "memory");
}
__device__ __forceinline__ void async_ld_b64(unsigned lds_off, const float* gaddr) {
    asm volatile("global_load_async_to_lds_b64 %0, %1, off"
                 :: "v"(lds_off), "v"(gaddr) : "memory");
}
__device__ __forceinline__ void wait_asynccnt0() {
    asm volatile("s_wait_asynccnt 0" ::: "memory");
}

__global__ __launch_bounds__(NTHREADS)
void pm_diffusion_step(const float* __restrict__ in, float* __restrict__ out,
                       int H, int W) {
    __shared__ __attribute__((aligned(16))) float s_in[IN_H * IN_STRIDE];
    __shared__ __attribute__((aligned(16))) float s_fx[FL_H * FL_W];
    __shared__ __attribute__((aligned(16))) float s_fy[FL_H * FL_W];

    const int tid = threadIdx.x;
    const int ox  = blockIdx.x * TILE_X;
    const int oy  = blockIdx.y * TILE_Y;
    const long long plane = (long long)blockIdx.z * H * W;
    const float* __restrict__ src = in  + plane;
    float* __restrict__       dst = out + plane;

    // ---- Phase 1a: middle 64 floats of each of the 36 rows: 16 x b128 per row
    for (int idx = tid; idx < IN_H * 16; idx += NTHREADS) {
        const int row = idx >> 4;
        const int c   = idx & 15;                 // 4-float chunk within row
        const int gy  = oy + row - 2;
        const int li  = SIN_IDX(row, 2 + 4 * c);  // 16B-aligned LDS slot
        if (gy >= 0 && gy < H) {
            async_ld_b128((unsigned)(uintptr_t)&s_in[li],
                          src + (long long)gy * W + (ox + 4 * c));
        } else {
            *(float4*)&s_in[li] = make_float4(0.f, 0.f, 0.f, 0.f);
        }
    }
    // ---- Phase 1b: 2-float halo pairs at ox-2 / ox+64 (8B-aligned, never straddle)
    for (int idx = tid; idx < IN_H * 2; idx += NTHREADS) {
        const int row  = idx >> 1;
        const int side = idx & 1;
        const int gy   = oy + row - 2;
        const int gx   = side ? (ox + TILE_X) : (ox - 2);
        const int li   = SIN_IDX(row, side ? (TILE_X + 2) : 0);
        if (gy >= 0 && gy < H && gx >= 0 && gx + 1 < W) {
            async_ld_b64((unsigned)(uintptr_t)&s_in[li],
                         src + (long long)gy * W + gx);
        } else {
            *(float2*)&s_in[li] = make_float2(0.f, 0.f);   // zero 'SAME' padding
        }
    }
    wait_asynccnt0();
    __syncthreads();

    // ---- Phase 2: flux on 66x34 region (output tile + halo 1).
    // Flux outside the image must be ZERO (reference zero-pads the flux arrays).
    const float KAP2_INV = 400.0f;               // 1/kappa^2, kappa = 0.05
    for (int idx = tid; idx < FL_H * FL_W; idx += NTHREADS) {
        const int fy = idx / FL_W;
        const int fx = idx - fy * FL_W;
        const int gy = oy + fy - 1;
        const int gx = ox + fx - 1;
        float vx = 0.0f, vy = 0.0f;
        if (gy >= 0 && gy < H && gx >= 0 && gx < W) {
            // 3x3 neighborhood centered at input-local (fy+1, fx+1)
            const float* p = &s_in[SIN_IDX(fy, fx)];
            const float a00 = p[0],              a01 = p[1],                  a02 = p[2];
            const float a10 = p[IN_STRIDE],                                   a12 = p[IN_STRIDE + 2];
            const float a20 = p[2 * IN_STRIDE],  a21 = p[2 * IN_STRIDE + 1],  a22 = p[2 * IN_STRIDE + 2];
            const float gxv = (a02 - a00 + 2.0f * (a12 - a10) + a22 - a20) * 0.125f;
            const float gyv = (a20 - a00 + 2.0f * (a21 - a01) + a22 - a02) * 0.125f;
            const float s   = gxv * gxv + gyv * gyv + 1e-8f;
            const float c   = 1.0f / (1.0f + s * KAP2_INV);   // Perona-Malik g2
            vx = c * gxv;
            vy = c * gyv;
        }
        s_fx[idx] = vx;
        s_fy[idx] = vy;
    }
    __syncthreads();

    // ---- Phase 3: divergence + Euler update on 64x32 outputs
    for (int idx = tid; idx < TILE_Y * TILE_X; idx += NTHREADS) {
        const int ty = idx / TILE_X;
        const int tx = idx - ty * TILE_X;
        const int fy = ty + 1;                   // flux-local center
        const int fx = tx + 1;
        const float* px = &s_fx[(fy - 1) * FL_W + (fx - 1)];
        const float* py = &s_fy[(fy - 1) * FL_W + (fx - 1)];
        const float dx = (px[2] - px[0]
                        + 2.0f * (px[FL_W + 2] - px[FL_W])
                        + px[2 * FL_W + 2] - px[2 * FL_W]) * 0.125f;
        const float dy = (py[2 * FL_W] - py[0]
                        + 2.0f * (py[2 * FL_W + 1] - py[1])
                        + py[2 * FL_W + 2] - py[2]) * 0.125f;
        const float v = s_in[SIN_IDX(ty + 2, tx + 2)] + 0.25f * (dx + dy);
        dst[(long long)(oy + ty) * W + (ox + tx)] = v;
    }
}

extern "C" void kernel_launch(void* const* d_in, const int* in_sizes, int n_in,
                              void* d_out, int out_size, void* d_ws, size_t ws_size,
                              hipStream_t stream) {
    const int N = 16, H = 1024, W = 1024;        // reference shape (16,1,1024,1024)
    const float* img = (const float*)d_in[0];
    float* out = (float*)d_out;
    float* tmp = (float*)d_ws;                   // needs one 64 MB image plane

    dim3 grid(W / TILE_X, H / TILE_Y, N);
    dim3 block(NTHREADS);

    // 3 fused diffusion iterations, ping-pong out <-> ws, finish in d_out.
    pm_diffusion_step<<<grid, block, 0, stream>>>(img, out, H, W);
    pm_diffusion_step<<<grid, block, 0, stream>>>(out, tmp, H, W);
    pm_diffusion_step<<<grid, block, 0, stream>>>(tmp, out, H, W);
}